// CenterVLAD_19842748907679
// MI455X (gfx1250) — compile-verified
//
#include <hip/hip_runtime.h>
#include <math.h>

typedef float v2f __attribute__((ext_vector_type(2)));
typedef float v8f __attribute__((ext_vector_type(8)));

#define D_DIM   32
#define K_DIM   116
#define K_PAD   128
#define HW      1024
#define B_DIM   32
#define NCHUNK  8      // pixel chunks per batch in k2 (occupancy)
#define CHUNKPX (HW / NCHUNK)

// ---------------------------------------------------------------------------
// Kernel 1: s = x @ W  (16x16x4 f32 WMMA), softmax over K=116 -> a (padded 128)
// block = 128 threads (4 waves), each block handles 64 pixels (16 per wave)
// ---------------------------------------------------------------------------
__global__ void __launch_bounds__(128)
k1_assign(const float* __restrict__ x, const float* __restrict__ W,
          float* __restrict__ a)
{
    __shared__ float s_lds[64][132];   // 64 rows x 128 cols (+4 pad)

    const int tid  = threadIdx.x;
    const int wave = tid >> 5;
    const int lane = tid & 31;
    const int half = lane >> 4;        // 0: lanes 0-15, 1: lanes 16-31
    const int l15  = lane & 15;

    const int rowg = blockIdx.x * 64 + wave * 16;   // global pixel base for this wave

    // 8 cluster tiles of 16 columns each
    for (int t = 0; t < 8; ++t) {
        const int k0 = t * 16;
        v8f acc = {};
        #pragma unroll
        for (int p0 = 0; p0 < D_DIM; p0 += 4) {
            const int p = p0 + 2 * half;
            // A fragment: 16x4 tile of x   A[m][p] = x[rowg+m][p]
            const float* xp = x + (size_t)(rowg + l15) * D_DIM + p;
            v2f af; af.x = xp[0]; af.y = xp[1];
            // B fragment: 4x16 tile of W (zero-padded beyond K=116)
            const int kk = k0 + l15;
            v2f bf;
            if (kk < K_DIM) {
                bf.x = W[(size_t)(p + 0) * K_DIM + kk];
                bf.y = W[(size_t)(p + 1) * K_DIM + kk];
            } else { bf.x = 0.f; bf.y = 0.f; }
            acc = __builtin_amdgcn_wmma_f32_16x16x4_f32(
                      false, af, false, bf, (short)0, acc, false, false);
        }
        // D layout: vgpr r -> row (half*8 + r), col l15
        #pragma unroll
        for (int r = 0; r < 8; ++r)
            s_lds[wave * 16 + half * 8 + r][k0 + l15] = acc[r];
    }
    __syncthreads();

    // softmax over the 116 valid columns; threads 0..63 each own one row
    if (tid < 64) {
        float m = -1e30f;
        for (int k = 0; k < K_DIM; ++k) m = fmaxf(m, s_lds[tid][k]);
        float sum = 0.f;
        for (int k = 0; k < K_DIM; ++k) {
            float e = expf(s_lds[tid][k] - m);
            s_lds[tid][k] = e;
            sum += e;
        }
        const float inv = 1.0f / sum;
        float* arow = a + (size_t)(blockIdx.x * 64 + tid) * K_PAD;
        for (int k = 0; k < K_DIM; ++k) arow[k] = s_lds[tid][k] * inv;
        for (int k = K_DIM; k < K_PAD; ++k) arow[k] = 0.f;   // pad -> zero contribution
    }
}

// ---------------------------------------------------------------------------
// Kernel 2: per (batch, pixel-chunk) partial of
//     v_raw[0:32] = x_b^T @ a_b  (D x K over CHUNKPX pixels)
//     v_raw[32]   = column sums of a_b (ones-row trick)
// grid = 32*8 blocks, block = 256 threads (8 waves); wave w -> cluster tile w
// ---------------------------------------------------------------------------
__global__ void __launch_bounds__(256)
k2_aggregate(const float* __restrict__ x, const float* __restrict__ a,
             float* __restrict__ vpart)
{
    const int tid   = threadIdx.x;
    const int wave  = tid >> 5;
    const int lane  = tid & 31;
    const int half  = lane >> 4;
    const int l15   = lane & 15;
    const int b     = blockIdx.x >> 3;                 // batch
    const int chunk = blockIdx.x & (NCHUNK - 1);       // pixel chunk
    const int k0    = wave * 16;

    const float* xb = x + ((size_t)b * HW + (size_t)chunk * CHUNKPX) * D_DIM;
    const float* ab = a + ((size_t)b * HW + (size_t)chunk * CHUNKPX) * K_PAD;

    v8f acc0 = {}, acc1 = {}, acc2 = {};
    // ones-row tile: A[m][p] = (m == 32) -> lane 0 / lane 16 carry the 1.0
    const float one = (l15 == 0) ? 1.0f : 0.0f;
    v2f af2; af2.x = one; af2.y = one;

    for (int p0 = 0; p0 < CHUNKPX; p0 += 4) {
        const int p = p0 + 2 * half;
        // B fragment from a (4 pixels x 16 clusters)
        v2f bf;
        bf.x = ab[(size_t)(p + 0) * K_PAD + k0 + l15];
        bf.y = ab[(size_t)(p + 1) * K_PAD + k0 + l15];
        // A fragments from x^T: A[m][p] = x[p][m], m-tiles 0..15 and 16..31
        v2f af0, af1;
        af0.x = xb[(size_t)(p + 0) * D_DIM + l15];
        af0.y = xb[(size_t)(p + 1) * D_DIM + l15];
        af1.x = xb[(size_t)(p + 0) * D_DIM + 16 + l15];
        af1.y = xb[(size_t)(p + 1) * D_DIM + 16 + l15];

        acc0 = __builtin_amdgcn_wmma_f32_16x16x4_f32(false, af0, false, bf, (short)0, acc0, false, false);
        acc1 = __builtin_amdgcn_wmma_f32_16x16x4_f32(false, af1, false, bf, (short)0, acc1, false, false);
        acc2 = __builtin_amdgcn_wmma_f32_16x16x4_f32(false, af2, false, bf, (short)0, acc2, false, false);
    }

    float* vb = vpart + ((size_t)b * NCHUNK + chunk) * 48 * K_PAD;
    const int col = k0 + l15;
    #pragma unroll
    for (int r = 0; r < 8; ++r) {
        vb[(size_t)( 0 + half * 8 + r) * K_PAD + col] = acc0[r];
        vb[(size_t)(16 + half * 8 + r) * K_PAD + col] = acc1[r];
        vb[(size_t)(32 + half * 8 + r) * K_PAD + col] = acc2[r];   // row 32 = asum
    }
}

// ---------------------------------------------------------------------------
// Kernel 3: sum partials; v = v_raw + asum*C; intra-norm over D; global L2
// grid = 32 batches, block = 128 threads (thread k = cluster column)
// ---------------------------------------------------------------------------
__global__ void __launch_bounds__(128)
k3_normalize(const float* __restrict__ vpart, const float* __restrict__ C,
             float* __restrict__ out)
{
    __shared__ float red[128];
    const int b = blockIdx.x;
    const int k = threadIdx.x;
    const float* vb = vpart + (size_t)b * NCHUNK * 48 * K_PAD;

    float col[D_DIM];
    float colsq = 0.f;
    if (k < K_DIM) {
        float asum = 0.f;
        #pragma unroll
        for (int c = 0; c < NCHUNK; ++c)
            asum += vb[(size_t)c * 48 * K_PAD + (size_t)32 * K_PAD + k];
        #pragma unroll
        for (int d = 0; d < D_DIM; ++d) {
            float val = 0.f;
            #pragma unroll
            for (int c = 0; c < NCHUNK; ++c)
                val += vb[(size_t)c * 48 * K_PAD + (size_t)d * K_PAD + k];
            val += asum * C[(size_t)d * K_DIM + k];
            col[d] = val;
            colsq += val * val;
        }
    }
    const float cinv = 1.0f / sqrtf(colsq + 1e-12f);
    red[k] = (k < K_DIM) ? colsq * cinv * cinv : 0.f;   // ||normalized col||^2
    __syncthreads();
    for (int s = 64; s > 0; s >>= 1) {
        if (k < s) red[k] += red[k + s];
        __syncthreads();
    }
    const float ginv = 1.0f / sqrtf(red[0] + 1e-12f);

    if (k < K_DIM) {
        const float scale = cinv * ginv;
        float* ob = out + (size_t)b * (D_DIM * K_DIM);
        #pragma unroll
        for (int d = 0; d < D_DIM; ++d)
            ob[(size_t)d * K_DIM + k] = col[d] * scale;
    }
}

// ---------------------------------------------------------------------------
extern "C" void kernel_launch(void* const* d_in, const int* in_sizes, int n_in,
                              void* d_out, int out_size, void* d_ws, size_t ws_size,
                              hipStream_t stream)
{
    const float* x = (const float*)d_in[0];   // (32,32,32,32)
    const float* W = (const float*)d_in[1];   // (32,116)
    const float* C = (const float*)d_in[2];   // (32,116)
    float* out = (float*)d_out;               // (32, 3712)

    float* a     = (float*)d_ws;                                  // 32768 x 128 f32 (16 MB)
    float* vpart = (float*)((char*)d_ws +
                   (size_t)B_DIM * HW * K_PAD * sizeof(float));   // 32 x 8 x 48 x 128 (6.3 MB)

    k1_assign   <<<(B_DIM * HW) / 64,  128, 0, stream>>>(x, W, a);
    k2_aggregate<<<B_DIM * NCHUNK,     256, 0, stream>>>(x, a, vpart);
    k3_normalize<<<B_DIM,              128, 0, stream>>>(vpart, C, out);
}